// EntropyBottleneck_34531537060507
// MI455X (gfx1250) — compile-verified
//
#include <hip/hip_runtime.h>
#include <cstdint>

#define LBND 1e-9f

// ---------------- gfx1250 hardware transcendentals ----------------
__device__ __forceinline__ float htanh(float x) {
    float r;
    asm("v_tanh_f32 %0, %1" : "=v"(r) : "v"(x));
    return r;
}
__device__ __forceinline__ float hexp2(float x) {
    float r;
    asm("v_exp_f32 %0, %1" : "=v"(r) : "v"(x));
    return r;
}
__device__ __forceinline__ float hlog2(float x) {
    float r;
    asm("v_log_f32 %0, %1" : "=v"(r) : "v"(x));
    return r;
}
__device__ __forceinline__ float hrcp(float x) {
    float r;
    asm("v_rcp_f32 %0, %1" : "=v"(r) : "v"(x));
    return r;
}
__device__ __forceinline__ float hsigmoid(float x) {
    // sigmoid(x) = 1 / (1 + 2^(-x*log2(e)))
    return hrcp(1.0f + hexp2(x * -1.4426950408889634f));
}
__device__ __forceinline__ float hsoftplus(float x) {
    // softplus(x) = max(x,0) + ln2 * log2(1 + 2^(-|x|*log2(e)))
    float t = hexp2(fabsf(x) * -1.4426950408889634f);
    return fmaxf(x, 0.0f) + 0.6931471805599453f * hlog2(1.0f + t);
}

// ---------------- per-channel parameter pack ----------------
// Layout per channel (64 floats):
//  [0..2]   w0 = softplus(matrix0)
//  [3..5]   b0L = bias0 - 0.5*w0          (lower-pass layer-0 bias, folds u-0.5)
//  [6..8]   t0 = tanh(factor0)
//  [9..17]  w1 (row-major)   [18..20] b1   [21..23] t1
//  [24..32] w2               [33..35] b2   [36..38] t2
//  [39..47] w3               [48..50] b3   [51..53] t3
//  [54..56] w4               [57]     b4
//  [58..60] b0U = bias0 + 0.5*w0          (upper-pass layer-0 bias, folds u+0.5)
__global__ void eb_prep(const float* __restrict__ m0, const float* __restrict__ b0, const float* __restrict__ f0,
                        const float* __restrict__ m1, const float* __restrict__ b1, const float* __restrict__ f1,
                        const float* __restrict__ m2, const float* __restrict__ b2, const float* __restrict__ f2,
                        const float* __restrict__ m3, const float* __restrict__ b3, const float* __restrict__ f3,
                        const float* __restrict__ m4, const float* __restrict__ b4,
                        float* __restrict__ cp) {
    int c = blockIdx.x * blockDim.x + threadIdx.x;
    if (c >= 192) return;
    float* o = cp + (size_t)c * 64;
    // layer 0 (1 -> 3) with +-0.5 folded into two bias sets
    #pragma unroll
    for (int r = 0; r < 3; ++r) {
        float w  = hsoftplus(m0[c * 3 + r]);
        float b  = b0[c * 3 + r];
        o[r]      = w;
        o[3 + r]  = fmaf(-0.5f, w, b);   // lower bias
        o[58 + r] = fmaf(0.5f, w, b);    // upper bias
        o[6 + r]  = htanh(f0[c * 3 + r]);
    }
    const float* Ms[3] = {m1, m2, m3};
    const float* Bs[3] = {b1, b2, b3};
    const float* Fs[3] = {f1, f2, f3};
    #pragma unroll
    for (int j = 0; j < 3; ++j) {
        int L = 9 + 15 * j;
        #pragma unroll
        for (int k = 0; k < 9; ++k) o[L + k] = hsoftplus(Ms[j][c * 9 + k]);
        #pragma unroll
        for (int r = 0; r < 3; ++r) {
            o[L + 9 + r]  = Bs[j][c * 3 + r];
            o[L + 12 + r] = htanh(Fs[j][c * 3 + r]);
        }
    }
    // layer 4 (3 -> 1)
    #pragma unroll
    for (int k = 0; k < 3; ++k) o[54 + k] = hsoftplus(m4[c * 3 + k]);
    o[57] = b4[c];
    o[61] = 0.0f; o[62] = 0.0f; o[63] = 0.0f;
}

// ---------------- main elementwise MLP ----------------
// BO = 3 (lower pass, bias pre-shifted by -0.5*w0) or 58 (upper pass, +0.5*w0)
template <int BO>
__device__ __forceinline__ float eval_mlp(float u, const float p[61]) {
    // layer 0 (1->3) + gate
    float a0 = fmaf(p[0], u, p[BO + 0]);
    float a1 = fmaf(p[1], u, p[BO + 1]);
    float a2 = fmaf(p[2], u, p[BO + 2]);
    a0 = fmaf(p[6], htanh(a0), a0);
    a1 = fmaf(p[7], htanh(a1), a1);
    a2 = fmaf(p[8], htanh(a2), a2);
    // layers 1..3 (3->3) + gate
    #pragma unroll
    for (int L = 9; L <= 39; L += 15) {
        float h0 = fmaf(p[L + 0], a0, fmaf(p[L + 1], a1, fmaf(p[L + 2], a2, p[L + 9])));
        float h1 = fmaf(p[L + 3], a0, fmaf(p[L + 4], a1, fmaf(p[L + 5], a2, p[L + 10])));
        float h2 = fmaf(p[L + 6], a0, fmaf(p[L + 7], a1, fmaf(p[L + 8], a2, p[L + 11])));
        a0 = fmaf(p[L + 12], htanh(h0), h0);
        a1 = fmaf(p[L + 13], htanh(h1), h1);
        a2 = fmaf(p[L + 14], htanh(h2), h2);
    }
    // layer 4 (3->1)
    return fmaf(p[54], a0, fmaf(p[55], a1, fmaf(p[56], a2, p[57])));
}

__global__ __launch_bounds__(256) void eb_main(const float* __restrict__ x,
                                               const float* __restrict__ nz,
                                               const float* __restrict__ cp,
                                               float* __restrict__ outp,
                                               float* __restrict__ like) {
    __shared__ float4 sx[2][256];
    __shared__ float4 sn[2][256];
    const int tid   = threadIdx.x;
    const int plane = blockIdx.x;            // B*C = 3072 planes of H*W = 4096
    const int c     = plane % 192;
    const float* __restrict__ pc = cp + (size_t)c * 64;
    // Per-channel constants are block-uniform: pin them in SGPRs.
    float p[61];
    #pragma unroll
    for (int i = 0; i < 61; ++i) p[i] = __builtin_amdgcn_readfirstlane(pc[i]);

    const size_t pbase = (size_t)plane * 4096;

    auto stage = [&](int k, int buf) {
        const size_t off = pbase + (size_t)k * 1024 + (size_t)tid * 4;
        unsigned lx = (unsigned)(uintptr_t)(&sx[buf][tid]);
        unsigned ln = (unsigned)(uintptr_t)(&sn[buf][tid]);
        unsigned long long gx = (unsigned long long)(uintptr_t)(x + off);
        unsigned long long gn = (unsigned long long)(uintptr_t)(nz + off);
        // gfx1250 async copy: global -> LDS, tracked by ASYNCcnt
        asm volatile("global_load_async_to_lds_b128 %0, %1, off" :: "v"(lx), "v"(gx) : "memory");
        asm volatile("global_load_async_to_lds_b128 %0, %1, off" :: "v"(ln), "v"(gn) : "memory");
    };

    stage(0, 0);
    #pragma unroll
    for (int k = 0; k < 4; ++k) {
        if (k < 3) {
            stage(k + 1, (k + 1) & 1);
            // async loads complete in order: cnt<=2 means stage k's 2 loads landed
            asm volatile("s_wait_asynccnt 2" ::: "memory");
        } else {
            asm volatile("s_wait_asynccnt 0" ::: "memory");
        }
        float4 xv = sx[k & 1][tid];
        float4 nv = sn[k & 1][tid];
        const float* xs = (const float*)&xv;
        const float* ns = (const float*)&nv;
        float4 uo, lk;
        float* uop = (float*)&uo;
        float* lkp = (float*)&lk;
        #pragma unroll
        for (int i = 0; i < 4; ++i) {
            float u  = xs[i] + ns[i];
            float lo = eval_mlp<3>(u, p);    // logits at u - 0.5 (bias-folded)
            float hi = eval_mlp<58>(u, p);   // logits at u + 0.5 (bias-folded)
            float s  = (lo + hi > 0.0f) ? -1.0f : 1.0f;   // sign trick for stability
            float d  = fabsf(hsigmoid(s * hi) - hsigmoid(s * lo));
            uop[i] = u;
            lkp[i] = fmaxf(d, LBND);
        }
        const size_t off = pbase + (size_t)k * 1024 + (size_t)tid * 4;
        *(float4*)(outp + off) = uo;
        *(float4*)(like + off) = lk;
    }
}

// ---------------- host launcher ----------------
extern "C" void kernel_launch(void* const* d_in, const int* in_sizes, int n_in,
                              void* d_out, int out_size, void* d_ws, size_t ws_size,
                              hipStream_t stream) {
    const float* x  = (const float*)d_in[0];
    const float* nz = (const float*)d_in[1];

    const float* mats[5];
    const float* bis[5];
    const float* fac[4];
    // dict order: m0,b0,f0, m1,b1,f1, ..., m4,b4  (in_sizes[3]==576)
    // signature order: m0..m4, b0..b4, f0..f3     (in_sizes[3]==1728)
    bool signature_order = (n_in >= 4 && in_sizes[3] == 1728);
    if (signature_order) {
        for (int i = 0; i < 5; ++i) mats[i] = (const float*)d_in[2 + i];
        for (int i = 0; i < 5; ++i) bis[i]  = (const float*)d_in[7 + i];
        for (int i = 0; i < 4; ++i) fac[i]  = (const float*)d_in[12 + i];
    } else {
        int idx = 2;
        for (int i = 0; i < 5; ++i) {
            mats[i] = (const float*)d_in[idx++];
            bis[i]  = (const float*)d_in[idx++];
            if (i < 4) fac[i] = (const float*)d_in[idx++];
        }
    }

    float* cp = (float*)d_ws;   // 192 channels * 64 floats = 48 KB

    eb_prep<<<6, 32, 0, stream>>>(mats[0], bis[0], fac[0],
                                  mats[1], bis[1], fac[1],
                                  mats[2], bis[2], fac[2],
                                  mats[3], bis[3], fac[3],
                                  mats[4], bis[4], cp);

    const int NT = 16 * 192 * 64 * 64;  // 12,582,912
    float* outp = (float*)d_out;
    float* like = outp + NT;
    eb_main<<<3072, 256, 0, stream>>>(x, nz, cp, outp, like);
}